// GCN_55920474193961
// MI455X (gfx1250) — compile-verified
//
#include <hip/hip_runtime.h>

// ---------------------------------------------------------------------------
// 2-layer GCN for MI455X (gfx1250, wave32).
//   h  = x @ W1                      (WMMA f32 16x16x4; W1 staged in LDS)
//   a1 = scatter-add(norm * h[src]) + selfloop   (L2 f32 atomics)
//   g  = relu(a1+b1) @ W2            (WMMA f32 16x16x4, K=64, Ncols=10->16)
//   out= scatter-add(norm * g[src]) + selfloop + b2
// norm(s,d) = rsqrt(deg[s]) * rsqrt(deg[d]), deg includes self loops.
// ---------------------------------------------------------------------------

typedef __attribute__((ext_vector_type(2))) float v2f;
typedef __attribute__((ext_vector_type(8))) float v8f;

#define FIN 500
#define HID 64
#define NC  10

// LDS image of W1, transposed: ldsW[c*LDSK + r] = W1[r][c].
// Stride 502 (even, 502 mod 64 = 54) -> 16 distinct banks for the 16 columns
// read per half-wave, and 8-byte alignment for b64 reads (both terms even).
#define LDSK 502
#define LDS_BYTES (HID * LDSK * 4)

// ---------------- degree / normalization ----------------------------------
__global__ void k_deg_init(float* __restrict__ deg, int n) {
    int i = blockIdx.x * blockDim.x + threadIdx.x;
    if (i < n) deg[i] = 1.0f;               // self loop
}

__global__ void k_deg_edges(const int* __restrict__ dst, float* __restrict__ deg, int e) {
    int i = blockIdx.x * blockDim.x + threadIdx.x;
    if (i < e) atomicAdd(&deg[dst[i]], 1.0f);
}

__global__ void k_dis(const float* __restrict__ deg, float* __restrict__ dis, int n) {
    int i = blockIdx.x * blockDim.x + threadIdx.x;
    if (i < n) dis[i] = rsqrtf(deg[i]);
}

// ---------------- GEMM1: h[N,64] = x[N,500] @ W1[500,64] -------------------
// 256-thread block = 8 waves sharing one LDS copy of W1 (transposed).
// Each wave owns 2 row-tiles (32 rows); 8 WMMA accumulators cover 32x64.
// Inner loop per K-step: 2 global b64 (A) + 4 ds b64 (B) -> 8 WMMAs.
__global__ __launch_bounds__(256)
void k_gemm1(const float* __restrict__ x, const float* __restrict__ W1,
             float* __restrict__ h, int n) {
    extern __shared__ float ldsW[];

    const int tid  = threadIdx.x;
    const int wave = tid >> 5;
    const int lane = tid & 31;
    const int m    = lane & 15;          // A row within tile / B column within tile
    const int kk   = (lane >> 4) * 2;    // lanes 0-15 -> K0,K1; lanes 16-31 -> K2,K3

    // ---- cooperative stage: W1[500][64] -> LDS transposed [64][502] ----
    for (int i = tid; i < FIN * HID; i += 256) {
        int r = i >> 6;                  // K index
        int c = i & 63;                  // column
        ldsW[c * LDSK + r] = W1[i];
    }
    __syncthreads();

    const int rowbase = blockIdx.x * 256 + wave * 32;   // 2 tiles: rowbase, rowbase+16
    long r0 = rowbase + m;       if (r0 >= n) r0 = n - 1;   // clamped loads
    long r1 = rowbase + 16 + m;  if (r1 >= n) r1 = n - 1;
    const float* xrow0 = x + r0 * (long)FIN;
    const float* xrow1 = x + r1 * (long)FIN;

    const float* wc0 = ldsW + (m)      * LDSK;   // column strips for the 4 N-tiles
    const float* wc1 = ldsW + (m + 16) * LDSK;
    const float* wc2 = ldsW + (m + 32) * LDSK;
    const float* wc3 = ldsW + (m + 48) * LDSK;

    v8f c00 = {}, c01 = {}, c02 = {}, c03 = {};   // tile 0 (rows rowbase..+15)
    v8f c10 = {}, c11 = {}, c12 = {}, c13 = {};   // tile 1 (rows rowbase+16..+31)
    for (int k = 0; k < FIN; k += 4) {
        v2f a0 = *(const v2f*)&xrow0[k + kk];
        v2f a1 = *(const v2f*)&xrow1[k + kk];
        v2f b0 = *(const v2f*)&wc0[k + kk];
        v2f b1 = *(const v2f*)&wc1[k + kk];
        v2f b2 = *(const v2f*)&wc2[k + kk];
        v2f b3 = *(const v2f*)&wc3[k + kk];
        c00 = __builtin_amdgcn_wmma_f32_16x16x4_f32(false, a0, false, b0, (short)0, c00, false, false);
        c01 = __builtin_amdgcn_wmma_f32_16x16x4_f32(false, a0, false, b1, (short)0, c01, false, false);
        c02 = __builtin_amdgcn_wmma_f32_16x16x4_f32(false, a0, false, b2, (short)0, c02, false, false);
        c03 = __builtin_amdgcn_wmma_f32_16x16x4_f32(false, a0, false, b3, (short)0, c03, false, false);
        c10 = __builtin_amdgcn_wmma_f32_16x16x4_f32(false, a1, false, b0, (short)0, c10, false, false);
        c11 = __builtin_amdgcn_wmma_f32_16x16x4_f32(false, a1, false, b1, (short)0, c11, false, false);
        c12 = __builtin_amdgcn_wmma_f32_16x16x4_f32(false, a1, false, b2, (short)0, c12, false, false);
        c13 = __builtin_amdgcn_wmma_f32_16x16x4_f32(false, a1, false, b3, (short)0, c13, false, false);
    }

    // D layout: VGPR j -> M=j (lanes 0-15) / M=8+j (lanes 16-31); N = lane&15
    const int mh = (lane >> 4) * 8;
    for (int j = 0; j < 8; ++j) {
        int row = rowbase + mh + j;
        if (row < n) {
            float* hr = h + (long)row * HID;
            hr[m]      = c00[j];
            hr[m + 16] = c01[j];
            hr[m + 32] = c02[j];
            hr[m + 48] = c03[j];
        }
        int row2 = rowbase + 16 + mh + j;
        if (row2 < n) {
            float* hr = h + (long)row2 * HID;
            hr[m]      = c10[j];
            hr[m + 16] = c11[j];
            hr[m + 32] = c12[j];
            hr[m + 48] = c13[j];
        }
    }
}

// ---------------- aggregation 1 (64 features) ------------------------------
__global__ void k_agg1_self(const float* __restrict__ h, const float* __restrict__ dis,
                            float* __restrict__ hagg, int n) {
    long t = (long)blockIdx.x * blockDim.x + threadIdx.x;     // n*64 threads
    if (t >= (long)n * HID) return;
    int i = (int)(t >> 6);
    float s = dis[i];
    hagg[t] = h[t] * s * s;                                   // self-loop term
}

__global__ void k_agg1_edges(const int* __restrict__ src, const int* __restrict__ dst,
                             const float* __restrict__ h, const float* __restrict__ dis,
                             float* __restrict__ hagg, int e) {
    long t = (long)blockIdx.x * blockDim.x + threadIdx.x;     // e*32 threads
    int  ed = (int)(t >> 5);
    if (ed >= e) return;
    int f2 = ((int)t & 31) * 2;
    int s = src[ed], d = dst[ed];
    float nrm = dis[s] * dis[d];
    const float* hs = h    + (long)s * HID + f2;
    float*       od = hagg + (long)d * HID + f2;
    atomicAdd(&od[0], hs[0] * nrm);
    atomicAdd(&od[1], hs[1] * nrm);
}

// ---------------- GEMM2: g[N,16] = relu(hagg+b1)[N,64] @ W2[64,10] ---------
// bias + relu fused into the A-operand load; N padded 10 -> 16 with zeros.
__global__ __launch_bounds__(32)
void k_gemm2(const float* __restrict__ hagg, const float* __restrict__ b1,
             const float* __restrict__ W2, float* __restrict__ g, int n) {
    const int lane = threadIdx.x;
    const int row0 = blockIdx.x * 16;
    const int m    = lane & 15;
    const int kk   = (lane >> 4) * 2;

    long rowm = row0 + m; if (rowm >= n) rowm = n - 1;
    const float* ar = hagg + rowm * (long)HID;

    v8f c = {};
    for (int k = 0; k < HID; k += 4) {
        v2f a;
        a.x = fmaxf(ar[k + kk]     + b1[k + kk],     0.0f);
        a.y = fmaxf(ar[k + kk + 1] + b1[k + kk + 1], 0.0f);
        v2f b;
        b.x = (m < NC) ? W2[(long)(k + kk)     * NC + m] : 0.0f;
        b.y = (m < NC) ? W2[(long)(k + kk + 1) * NC + m] : 0.0f;
        c = __builtin_amdgcn_wmma_f32_16x16x4_f32(false, a, false, b, (short)0, c, false, false);
    }
    const int mh = (lane >> 4) * 8;
    for (int j = 0; j < 8; ++j) {
        int row = row0 + mh + j;
        if (row < n) g[(long)row * 16 + m] = c[j];   // padded stride 16
    }
}

// ---------------- aggregation 2 (10 outputs) -------------------------------
__global__ void k_agg2_self(const float* __restrict__ g, const float* __restrict__ dis,
                            const float* __restrict__ b2, float* __restrict__ out, int n) {
    long t = (long)blockIdx.x * blockDim.x + threadIdx.x;     // n*10 threads
    if (t >= (long)n * NC) return;
    int i = (int)(t / NC);
    int c = (int)(t - (long)i * NC);
    float s = dis[i];
    out[t] = g[(long)i * 16 + c] * s * s + b2[c];
}

__global__ void k_agg2_edges(const int* __restrict__ src, const int* __restrict__ dst,
                             const float* __restrict__ g, const float* __restrict__ dis,
                             float* __restrict__ out, int e) {
    long t = (long)blockIdx.x * blockDim.x + threadIdx.x;     // e*16 threads
    int  ed = (int)(t >> 4);
    if (ed >= e) return;
    int c = (int)t & 15;
    if (c >= NC) return;
    int s = src[ed], d = dst[ed];
    float nrm = dis[s] * dis[d];
    atomicAdd(&out[(long)d * NC + c], g[(long)s * 16 + c] * nrm);
}

// ---------------------------------------------------------------------------
static inline unsigned cdiv(long a, long b) { return (unsigned)((a + b - 1) / b); }

extern "C" void kernel_launch(void* const* d_in, const int* in_sizes, int n_in,
                              void* d_out, int out_size, void* d_ws, size_t ws_size,
                              hipStream_t stream) {
    const float* x  = (const float*)d_in[0];
    const int*   ei = (const int*)  d_in[1];
    const float* W1 = (const float*)d_in[2];
    const float* b1 = (const float*)d_in[3];
    const float* W2 = (const float*)d_in[4];
    const float* b2 = (const float*)d_in[5];
    float* out = (float*)d_out;

    const int E = in_sizes[1] / 2;          // edge_index is [2, E]
    const int N = in_sizes[0] / FIN;        // x is [N, 500]
    const int* src = ei;
    const int* dst = ei + E;

    // workspace layout (all f32): deg[N] dis[N] h[N*64] hagg[N*64] g[N*16]
    float* deg  = (float*)d_ws;
    float* dis  = deg  + N;
    float* h    = dis  + N;
    float* hagg = h    + (size_t)N * HID;
    float* g    = hagg + (size_t)N * HID;

    // normalization
    k_deg_init <<<cdiv(N, 256), 256, 0, stream>>>(deg, N);
    k_deg_edges<<<cdiv(E, 256), 256, 0, stream>>>(dst, deg, E);
    k_dis      <<<cdiv(N, 256), 256, 0, stream>>>(deg, dis, N);

    // layer 1: GEMM1 blocks cover 256 rows each (8 waves x 2 row-tiles)
    k_gemm1    <<<cdiv(N, 256), 256, LDS_BYTES, stream>>>(x, W1, h, N);
    k_agg1_self<<<cdiv((long)N * HID, 256), 256, 0, stream>>>(h, dis, hagg, N);
    k_agg1_edges<<<cdiv((long)E * 32, 256), 256, 0, stream>>>(src, dst, h, dis, hagg, E);

    // layer 2 (bias+relu fused into GEMM2 A-loads)
    k_gemm2    <<<cdiv(N, 16), 32, 0, stream>>>(hagg, b1, W2, g, N);
    k_agg2_self<<<cdiv((long)N * NC, 256), 256, 0, stream>>>(g, dis, b2, out, N);
    k_agg2_edges<<<cdiv((long)E * 16, 256), 256, 0, stream>>>(src, dst, g, dis, out, E);
}